// TemporalAttentionModule_75316546503031
// MI455X (gfx1250) — compile-verified
//
#include <hip/hip_runtime.h>
#include <hip/hip_bf16.h>
#include <math.h>

// Problem constants (from reference): T=128, B=1, H=W=32, F=512 -> L = 32*32*512 = 524288 per t
// h: [HID=128, HH*WW=1024], Wx: [512], Wh: [128]. Output: 128 floats.

typedef float v2f __attribute__((ext_vector_type(2)));
typedef float v8f __attribute__((ext_vector_type(8)));

#define T_STEPS   128
#define L_PER_T   524288      // 32*32*512 floats
#define NWG_PER_T 8
#define L_PER_WG  (L_PER_T / NWG_PER_T)   // 65536 floats
#define ITERS     (L_PER_WG / (256 * 4))  // 64 float4 loads per thread
#define HID       128
#define HSPATIAL  1024

// ---------------------------------------------------------------------------
// Kernel 1: bandwidth-bound streaming weighted reduction over x.
// acc[t,wg] = sum over this WG's slice of x[t] * Wx[f]; f = flat_off % 512 is
// constant per lane because the tile stride (1024 floats) is a multiple of F.
// ---------------------------------------------------------------------------
__global__ __launch_bounds__(256) void tam_xdot_kernel(
    const float* __restrict__ x, const float* __restrict__ Wx,
    float* __restrict__ partials)
{
    const int tid = threadIdx.x;
    const int wg  = blockIdx.x;   // 0..7
    const int t   = blockIdx.y;   // 0..127

    // Per-thread weight vector: f index = (tid*4) mod 512, fixed across iterations.
    const float4 w4 = *(const float4*)(Wx + ((tid * 4) & 511));

    const float* base = x + (size_t)t * L_PER_T + (size_t)wg * L_PER_WG + tid * 4;

    float acc = 0.0f;
#pragma unroll 8
    for (int it = 0; it < ITERS; ++it) {
        // Prefetch the stream ~16 tiles ahead (speculative; OOB is dropped).
        __builtin_prefetch(base + it * 1024 + 16 * 1024, 0, 0);
        float4 v = *(const float4*)(base + it * 1024);
        acc = fmaf(v.x, w4.x, acc);
        acc = fmaf(v.y, w4.y, acc);
        acc = fmaf(v.z, w4.z, acc);
        acc = fmaf(v.w, w4.w, acc);
    }

    __shared__ float red[256];
    red[tid] = acc;
    __syncthreads();
    for (int s = 128; s > 0; s >>= 1) {
        if (tid < s) red[tid] += red[tid + s];
        __syncthreads();
    }
    if (tid == 0) partials[t * NWG_PER_T + wg] = red[0];
}

// ---------------------------------------------------------------------------
// Kernel 2 (one 256-thread block = 8 wave32):
//  1) pooled_h row-sums of h[128][1024] via chained V_WMMA_F32_16X16X4_F32
//     with B = ones (D[m][*] = running row-sum). Wave w handles rows 16w..16w+15.
//     A layout per ISA: lanes 0-15 -> (M=lane, K=0,1), lanes 16-31 -> (M=lane-16, K=2,3).
//  2) att_h = (1/1024) * dot(rowsums, Wh)   (fixed-order LDS tree)
//  3) e[t] = partialsum[t]/1024 + att_h ; softmax over t ; write out[128].
// ---------------------------------------------------------------------------
__global__ __launch_bounds__(256) void tam_finalize_kernel(
    const float* __restrict__ h, const float* __restrict__ Wh,
    const float* __restrict__ partials, float* __restrict__ out)
{
    const int tid  = threadIdx.x;
    const int lane = tid & 31;
    const int wave = tid >> 5;            // 0..7 -> row tile 16*wave

    __shared__ float pooled[HID];
    __shared__ float red[256];

    // ---- 1) WMMA row-sum chain (EXEC all ones here: no divergence yet) ----
    const int rowInTile = lane & 15;
    const int row   = wave * 16 + rowInTile;
    const int kOff  = (lane >> 4) << 1;   // 0 for lanes 0-15, 2 for lanes 16-31
    const float* hrow = h + (size_t)row * HSPATIAL + kOff;

    v8f c = {};                           // zero accumulator
    v2f b; b.x = 1.0f; b.y = 1.0f;        // all-ones B matrix (layout-agnostic)
#pragma unroll 4
    for (int k = 0; k < HSPATIAL; k += 4) {
        float2 hv = *(const float2*)(hrow + k);
        v2f a; a.x = hv.x; a.y = hv.y;
        // D = A x ones + C  ==> per-row partial sums accumulate in C
        c = __builtin_amdgcn_wmma_f32_16x16x4_f32(
                /*neg_a=*/false, a, /*neg_b=*/false, b,
                /*c_mod=*/(short)0, c, /*reuse_a=*/false, /*reuse_b=*/false);
    }
    // D layout: VGPR j, lane n<16 -> D[j][n]; lane n>=16 -> D[j+8][n-16].
    // All columns equal the row sum; use column 0 (lane 0) and column 0 of
    // the upper half (lane 16).
    if (lane == 0) {
#pragma unroll
        for (int j = 0; j < 8; ++j) pooled[wave * 16 + j] = c[j];
    }
    if (lane == 16) {
#pragma unroll
        for (int j = 0; j < 8; ++j) pooled[wave * 16 + 8 + j] = c[j];
    }
    __syncthreads();

    // ---- 2) att_h = (1/1024) * sum_d pooled[d] * Wh[d] ----
    float v = 0.0f;
    if (tid < HID) v = pooled[tid] * Wh[tid];
    red[tid] = v;
    __syncthreads();
    for (int s = 128; s > 0; s >>= 1) {
        if (tid < s) red[tid] += red[tid + s];
        __syncthreads();
    }
    const float att_h = red[0] * (1.0f / 1024.0f);
    __syncthreads();

    // ---- 3) e[t] and softmax over T=128 ----
    float e = -INFINITY;
    if (tid < T_STEPS) {
        float s = 0.0f;
#pragma unroll
        for (int j = 0; j < NWG_PER_T; ++j) s += partials[tid * NWG_PER_T + j];
        e = s * (1.0f / 1024.0f) + att_h;
    }

    // max reduction
    red[tid] = e;
    __syncthreads();
    for (int s = 128; s > 0; s >>= 1) {
        if (tid < s) red[tid] = fmaxf(red[tid], red[tid + s]);
        __syncthreads();
    }
    const float m = red[0];
    __syncthreads();

    // sum of exp
    float ex = (tid < T_STEPS) ? expf(e - m) : 0.0f;
    red[tid] = ex;
    __syncthreads();
    for (int s = 128; s > 0; s >>= 1) {
        if (tid < s) red[tid] += red[tid + s];
        __syncthreads();
    }
    const float denom = red[0];

    if (tid < T_STEPS) out[tid] = ex / denom;
}

extern "C" void kernel_launch(void* const* d_in, const int* in_sizes, int n_in,
                              void* d_out, int out_size, void* d_ws, size_t ws_size,
                              hipStream_t stream) {
    const float* x  = (const float*)d_in[0];   // [128,1,32,32,512]
    const float* h  = (const float*)d_in[1];   // [1,128,32,32]
    const float* Wx = (const float*)d_in[2];   // [1,512]
    const float* Wh = (const float*)d_in[3];   // [1,128]
    float* out      = (float*)d_out;           // [128]
    float* partials = (float*)d_ws;            // 128*8 floats = 4 KB scratch

    dim3 grid1(NWG_PER_T, T_STEPS);
    tam_xdot_kernel<<<grid1, 256, 0, stream>>>(x, Wx, partials);
    tam_finalize_kernel<<<1, 256, 0, stream>>>(h, Wh, partials, out);
}